// QuantumLayer_1812476199025
// MI455X (gfx1250) — compile-verified
//
#include <hip/hip_runtime.h>

typedef __attribute__((ext_vector_type(2))) float v2f;
typedef __attribute__((ext_vector_type(4))) float v4f;
typedef __attribute__((ext_vector_type(8))) float v8f;

#define NQ 10
#define DIM 1024            // 2^10 statevector
#define WAVES_PER_BLOCK 8
#define THREADS (WAVES_PER_BLOCK * 32)
#define NGATES 40           // 2 layers x (10 Rot + 10 CRY)

// Branchless B-matrix entry for the <Z> readout GEMM:
// cols 0..5 = +/-1 from bit n of t (wire 9-n), col 6 = 1 (rowsum), cols 7..15 = 0.
static __device__ __forceinline__ float zsign(int n, int t) {
    float s = 1.0f - 2.0f * (float)((t >> n) & 1);
    s = (n == 6) ? 1.0f : s;
    s = (n > 6)  ? 0.0f : s;
    return s;
}

// ---------------------------------------------------------------------------
// Prep kernel: gate coefficients are sample-independent; compute the 40 gate
// matrices once (8 floats each) into d_ws so the 2048 simulation waves do no
// gate trig at all (uniform scalar loads instead).
// Gate gi = L*20 + q      (Rot on wire q, layer L)   -> m00r,m00i,m01r,m01i,m10r,m10i,m11r,m11i
//      gi = L*20 + 10 + e (CRY on edge e, layer L)   -> c,s,0,0,0,0,0,0
// ---------------------------------------------------------------------------
__global__ void prep_gates(const float* __restrict__ u3,
                           const float* __restrict__ ent,
                           float* __restrict__ gw)
{
    int gi = threadIdx.x;
    if (gi >= NGATES) return;
    int L = gi / 20;
    int r = gi % 20;
    float c8[8] = {0.f, 0.f, 0.f, 0.f, 0.f, 0.f, 0.f, 0.f};
    if (r < 10) {
        const float* g = &u3[(L * NQ + r) * 3];
        float phi = g[0], th = g[1], om = g[2];
        float ct = cosf(0.5f * th), st = sinf(0.5f * th);
        float a = 0.5f * (phi + om), b = 0.5f * (phi - om);
        float ca = cosf(a), sa = sinf(a), cb = cosf(b), sb = sinf(b);
        c8[0] =  ca * ct;  c8[1] = -sa * ct;   // m00
        c8[2] = -cb * st;  c8[3] = -sb * st;   // m01
        c8[4] =  cb * st;  c8[5] = -sb * st;   // m10
        c8[6] =  ca * ct;  c8[7] =  sa * ct;   // m11
    } else {
        float w = ent[L * NQ + (r - 10)] + 0.3f;
        c8[0] = cosf(0.5f * w);
        c8[1] = sinf(0.5f * w);
    }
#pragma unroll
    for (int k = 0; k < 8; ++k) gw[gi * 8 + k] = c8[k];
}

// ---------------------------------------------------------------------------
// Main kernel: one wave32 per sample; statevector as interleaved complex
// (float2) in LDS; all gate sweeps intra-wave (no workgroup barriers).
// ---------------------------------------------------------------------------
__global__ __launch_bounds__(THREADS)
void qsim_kernel(const float* __restrict__ x,     // (nsamples, 10)
                 const float* __restrict__ gw,    // (40, 8) gate coeffs
                 float* __restrict__ out,         // (nsamples, 10)
                 int nsamples)
{
    __shared__ v2f st[WAVES_PER_BLOCK][DIM];      // 8 * 1024 * 8B = 64 KB

    const int lane   = threadIdx.x & 31;
    const int wv     = threadIdx.x >> 5;
    const int sample = blockIdx.x * WAVES_PER_BLOCK + wv;
    if (sample >= nsamples) return;               // wave-uniform

    v2f* S = st[wv];
    const float PI = 3.14159265358979323846f;

    // ---- AngleEmbedding RY folded into closed-form product state --------
    // wire w <-> flat-index bit (9 - w); amp(i) = prod_w (bit? sin : cos)(theta_w/2)
    float cw[NQ], sw[NQ];
#pragma unroll
    for (int q = 0; q < NQ; ++q) {
        float xv = x[sample * NQ + q];
        xv = fminf(fmaxf(xv, 0.0f), 1.0f);
        float th2 = 0.5f * PI * xv;               // theta/2
        cw[q] = cosf(th2);
        sw[q] = sinf(th2);
    }
    float lowp = 1.0f;                            // low 5 index bits = lane
#pragma unroll
    for (int j = 0; j < 5; ++j)
        lowp *= ((lane >> j) & 1) ? sw[9 - j] : cw[9 - j];
#pragma unroll 4
    for (int t = 0; t < 32; ++t) {                // high 5 index bits = t
        float hp = 1.0f;
#pragma unroll
        for (int j = 0; j < 5; ++j)
            hp *= ((t >> j) & 1) ? sw[4 - j] : cw[4 - j];
        v2f a; a.x = lowp * hp; a.y = 0.0f;
        S[t * 32 + lane] = a;
    }
    __builtin_amdgcn_wave_barrier();

    // ---- 2 x (10 Rot + 10 ring CRY) gate sweeps -------------------------
    for (int L = 0; L < 2; ++L) {
        // Rot gates (dense 2x2 complex), wire q -> bit (9-q)
        for (int q = 0; q < NQ; ++q) {
            const float* g = &gw[(L * 20 + q) * 8];
            float m00r = g[0], m00i = g[1], m01r = g[2], m01i = g[3];
            float m10r = g[4], m10i = g[5], m11r = g[6], m11i = g[7];
            const int bp = 9 - q;
            const unsigned lowmask = (1u << bp) - 1u;
#pragma unroll 2
            for (int it = 0; it < 16; ++it) {
                unsigned p  = (unsigned)(it * 32 + lane);        // [0,512)
                unsigned i0 = ((p >> bp) << (bp + 1)) | (p & lowmask);
                unsigned i1 = i0 | (1u << bp);
                v2f a0 = S[i0];
                v2f a1 = S[i1];
                v2f n0, n1;
                n0.x = m00r * a0.x - m00i * a0.y + m01r * a1.x - m01i * a1.y;
                n0.y = m00r * a0.y + m00i * a0.x + m01r * a1.y + m01i * a1.x;
                n1.x = m10r * a0.x - m10i * a0.y + m11r * a1.x - m11i * a1.y;
                n1.y = m10r * a0.y + m10i * a0.x + m11r * a1.y + m11i * a1.x;
                S[i0] = n0;
                S[i1] = n1;
            }
            __builtin_amdgcn_wave_barrier();
        }
        // CRY on ring edges (e, e+1 mod 10): control bit (9-e), target bit (9-v)
        for (int e = 0; e < NQ; ++e) {
            int v = (e + 1) % NQ;
            const float* g = &gw[(L * 20 + 10 + e) * 8];
            float c = g[0], s = g[1];
            const int bc = 9 - e, bt = 9 - v;
            const int blo = bc < bt ? bc : bt;
            const int bhi = bc < bt ? bt : bc;
            const unsigned mlo = (1u << blo) - 1u;
            const unsigned mhi = (1u << bhi) - 1u;
#pragma unroll 2
            for (int it = 0; it < 8; ++it) {
                unsigned p  = (unsigned)(it * 32 + lane);        // [0,256)
                unsigned t1 = ((p  >> blo) << (blo + 1)) | (p  & mlo);
                unsigned ib = ((t1 >> bhi) << (bhi + 1)) | (t1 & mhi);
                unsigned i0 = ib | (1u << bc);                   // |c=1,t=0>
                unsigned i1 = i0 | (1u << bt);                   // |c=1,t=1>
                v2f a0 = S[i0];
                v2f a1 = S[i1];
                v2f n0, n1;
                n0.x = c * a0.x - s * a1.x;  n0.y = c * a0.y - s * a1.y;
                n1.x = s * a0.x + c * a1.x;  n1.y = s * a0.y + c * a1.y;
                S[i0] = n0;
                S[i1] = n1;
            }
            __builtin_amdgcn_wave_barrier();
        }
    }

    // ---- <Z_q> readout as 16x64 . 64x16 GEMM via V_WMMA_F32_16X16X4_F32 -
    // i = m*64 + t: m = bits 6..9 (wires 3..0), t = bits 0..5 (wires 9..4).
    // D[m][n<6] = sum_t p[m*64+t]*sign_n(t);  D[m][6] = rowsum[m].
    v8f acc = {};
    const int half = lane >> 4;
    const int mn   = lane & 15;                  // A row / B col for this lane
#pragma unroll 4
    for (int kb = 0; kb < 16; ++kb) {
        int t0  = kb * 4 + 2 * half;             // this lane's 2 K values
        int idx = mn * 64 + t0;
        v4f amp2 = *(const v4f*)&S[idx];         // re0, im0, re1, im1
        v2f A, B;
        A.x = amp2.x * amp2.x + amp2.y * amp2.y; // p[idx]
        A.y = amp2.z * amp2.z + amp2.w * amp2.w; // p[idx+1]
        B.x = zsign(mn, t0);
        B.y = zsign(mn, t0 + 1);
        acc = __builtin_amdgcn_wmma_f32_16x16x4_f32(
                  false, A, false, B, (short)0, acc, false, false);
    }

    // C/D layout: reg v, lanes 0-15 -> (m=v, n=lane); lanes 16-31 -> (m=v+8).
    float colsum = acc[0] + acc[1] + acc[2] + acc[3]
                 + acc[4] + acc[5] + acc[6] + acc[7];
    float colfull = colsum + __shfl_xor(colsum, 16, 32);  // z(wire 9-n), n<6

    float zh[4];                                 // wires 0..3 from rowsum col 6
#pragma unroll
    for (int q = 0; q < 4; ++q) {
        float zq = 0.0f;
#pragma unroll
        for (int vreg = 0; vreg < 8; ++vreg) {
            int m = vreg + 8 * half;
            zq += (((m >> (3 - q)) & 1) ? -1.0f : 1.0f) * acc[vreg];
        }
        zh[q] = zq + __shfl_xor(zq, 16, 32);
    }

    if (lane < 6)
        out[sample * NQ + (9 - lane)] = colfull;
    if (lane == 6) {
#pragma unroll
        for (int q = 0; q < 4; ++q)
            out[sample * NQ + q] = zh[q];
    }
}

extern "C" void kernel_launch(void* const* d_in, const int* in_sizes, int n_in,
                              void* d_out, int out_size, void* d_ws, size_t ws_size,
                              hipStream_t stream) {
    const float* x   = (const float*)d_in[0];   // (8,256,10) f32
    const float* u3  = (const float*)d_in[1];   // (60,) f32
    const float* ent = (const float*)d_in[2];   // (20,) f32
    float* out = (float*)d_out;                 // (8,256,10) f32
    float* gw  = (float*)d_ws;                  // 40*8 floats of scratch
    (void)n_in; (void)out_size; (void)ws_size;

    prep_gates<<<1, 64, 0, stream>>>(u3, ent, gw);

    int nsamples = in_sizes[0] / NQ;            // 2048
    int blocks   = (nsamples + WAVES_PER_BLOCK - 1) / WAVES_PER_BLOCK;
    qsim_kernel<<<blocks, THREADS, 0, stream>>>(x, gw, out, nsamples);
}